// MyNeuralMemory_43550968381809
// MI455X (gfx1250) — compile-verified
//
#include <hip/hip_runtime.h>
#include <math.h>

// ---------------------------------------------------------------------------
// titans neural-memory store step for MI455X (gfx1250, wave32, WMMA)
//   Phase 1: per-token grads via V_WMMA_F32_16X16X4_F32 (fp32 matrix path),
//            rank-1 factorization + softclamp scales -> ~5MB factor workspace
//   Phase 2: 257-step gated scan, surprise reconstructed from rank-1 factors,
//            404MB streamed with nontemporal B128 stores (bandwidth bound).
// ---------------------------------------------------------------------------

#define TOKENS 1024          // B*N = 4*256
#define NTOK   256           // tokens per batch
#define BATCH  4
#define DIMV   64
#define HIDV   256
#define DTOT   98368         // 64*256 + 256*256 + 256*64 + 64
#define NP1    257
#define EPSV   1.1920929e-07f
#define GATE   1.0e-4f

typedef __attribute__((ext_vector_type(2))) float v2f;
typedef __attribute__((ext_vector_type(4))) float v4f;
typedef __attribute__((ext_vector_type(8))) float v8f;

// workspace layout (float offsets); total ~4.98 MB
#define OFF_XF   0
#define OFF_A0F  (OFF_XF  + TOKENS*DIMV)
#define OFF_A1F  (OFF_A0F + TOKENS*HIDV)
#define OFF_D0F  (OFF_A1F + TOKENS*HIDV)
#define OFF_D1F  (OFF_D0F + TOKENS*HIDV)
#define OFF_D2F  (OFF_D1F + TOKENS*HIDV)
#define OFF_GNF  (OFF_D2F + TOKENS*DIMV)
#define OFF_ONE  (OFF_GNF + TOKENS*DIMV)

__device__ __forceinline__ float wred(float v) {
  v += __shfl_xor(v, 16, 32);
  v += __shfl_xor(v, 8, 32);
  v += __shfl_xor(v, 4, 32);
  v += __shfl_xor(v, 2, 32);
  v += __shfl_xor(v, 1, 32);
  return v;
}

__device__ __forceinline__ float gelu_f(float x) {
  return 0.5f * x * (1.0f + erff(x * 0.70710678118654752f));
}
__device__ __forceinline__ float gelu_g(float x) {
  return 0.5f * (1.0f + erff(x * 0.70710678118654752f)) +
         x * 0.39894228040143267f * expf(-0.5f * x * x);
}
// softclamp_grad_norm scale (negated sign folded by caller)
__device__ __forceinline__ float sc_scale(float n) {
  const float h = 2.5f;  // MAX_GRAD_NORM/2
  float c = tanhf(n * (1.0f / h)) * h + h;
  return c / fmaxf(n, 1e-30f);
}

// one 16x16 fp32 WMMA tile: D(16xN-tile) += A(16xK) * B(KxN) using 16x16x4 steps
// A from LDS (row-major, stride lda). B from global: transB? B[k][n]=Bg[n*ldb+k]
//                                                   else    B[k][n]=Bg[k*ldb+n]
__device__ __forceinline__ void gemm_tile(const float* __restrict__ A, int lda,
                                          const float* __restrict__ Bg, int ldb,
                                          bool transB, int K,
                                          float* __restrict__ D, int ldd,
                                          int n0, int lane) {
  v8f c = {};
  const int m  = lane & 15;
  const int kh = (lane >> 4) << 1;         // 0 or 2 (K half per ISA A layout)
  const int nc = n0 + (lane & 15);
  for (int kk = 0; kk < K; kk += 4) {
    const int bk = kk + kh;
    v2f a, b;
    a.x = A[m * lda + bk];
    a.y = A[m * lda + bk + 1];
    if (!transB) {
      b.x = Bg[(size_t)bk * ldb + nc];
      b.y = Bg[(size_t)(bk + 1) * ldb + nc];
    } else {
      b.x = Bg[(size_t)nc * ldb + bk];
      b.y = Bg[(size_t)nc * ldb + bk + 1];
    }
    c = __builtin_amdgcn_wmma_f32_16x16x4_f32(false, a, false, b,
                                              (short)0, c, false, false);
  }
  const int rb = (lane >> 4) * 8;  // lanes 16-31 hold rows M=8..15
#pragma unroll
  for (int v = 0; v < 8; ++v)
    D[(size_t)(v + rb) * ldd + n0 + (lane & 15)] = c[v];
}

// ---------------------------------------------------------------------------
// Phase 1: 64 blocks x 512 threads (16 waves). Block handles 16 tokens.
// Wave w is N-tile [16w,16w+16) for GEMMs and owner of token w for vector ops.
// ---------------------------------------------------------------------------
__launch_bounds__(512)
__global__ void titans_phase1(const float* __restrict__ seq,
                              const float* __restrict__ w0,
                              const float* __restrict__ w1,
                              const float* __restrict__ w2,
                              const float* __restrict__ norm_w,
                              const float* __restrict__ store_nw,
                              float* __restrict__ ws) {
  const int tid  = threadIdx.x;
  const int wave = tid >> 5;
  const int lane = tid & 31;
  const int T0   = blockIdx.x * 16;

  if (blockIdx.x == 0 && tid == 0) ws[OFF_ONE] = 1.0f;  // stride-0 unit factor

  const float* keys = seq + (size_t)TOKENS * DIMV;       // seq[1]
  const float* vals = seq + (size_t)2 * TOKENS * DIMV;   // seq[2]

  float* Xf  = ws + OFF_XF;
  float* A0f = ws + OFF_A0F;
  float* A1f = ws + OFF_A1F;
  float* D0f = ws + OFF_D0F;
  float* D1f = ws + OFF_D1F;
  float* D2f = ws + OFF_D2F;
  float* GNf = ws + OFF_GNF;

  constexpr int WB = 260;  // padded stride for 256-wide rows (bank stride 4)
  constexpr int WS = 68;   // padded stride for 64-wide rows
  __shared__ float B1[16 * WB];  // h0 -> gelu'(h0)
  __shared__ float B2[16 * WB];  // a0 -> a1 -> dA1/dH1
  __shared__ float B3[16 * WB];  // h1 -> gelu'(h1) -> dA0
  __shared__ float sX[16 * WS];  // rmsnorm(keys)
  __shared__ float sY[16 * WS];  // rmsnorm(values)
  __shared__ float sT[16 * WS];  // h2 -> dh2

  // ---- P1: store-norm of keys/values, ||x|| ----
  float nx;
  {
    const int t = wave;
    const size_t tok = (size_t)(T0 + t);
    float k0 = keys[tok * 64 + lane], k1 = keys[tok * 64 + lane + 32];
    float u0 = vals[tok * 64 + lane], u1 = vals[tok * 64 + lane + 32];
    float rk = rsqrtf(wred(k0 * k0 + k1 * k1) * (1.0f / 64.0f) + EPSV);
    float rv = rsqrtf(wred(u0 * u0 + u1 * u1) * (1.0f / 64.0f) + EPSV);
    float sw0 = store_nw[lane], sw1 = store_nw[lane + 32];
    float x0 = k0 * rk * sw0, x1 = k1 * rk * sw1;
    float y0 = u0 * rv * sw0, y1 = u1 * rv * sw1;
    sX[t * WS + lane] = x0; sX[t * WS + lane + 32] = x1;
    sY[t * WS + lane] = y0; sY[t * WS + lane + 32] = y1;
    Xf[tok * 64 + lane] = x0; Xf[tok * 64 + lane + 32] = x1;
    nx = sqrtf(wred(x0 * x0 + x1 * x1));
  }
  __syncthreads();

  // ---- P2: H0 = X @ W0  (16x256, K=64) ----
  gemm_tile(sX, WS, w0, HIDV, false, DIMV, B1, WB, wave * 16, lane);
  __syncthreads();

  // ---- P3: a0 = gelu(h0), gp0 in place; ||a0|| ----
  for (int idx = tid; idx < 16 * HIDV; idx += 512) {
    int t = idx >> 8, j = idx & 255;
    float h = B1[t * WB + j];
    float a = gelu_f(h);
    B2[t * WB + j] = a;
    B1[t * WB + j] = gelu_g(h);
    A0f[(size_t)(T0 + t) * HIDV + j] = a;
  }
  __syncthreads();
  float na0;
  {
    float s = 0.f;
#pragma unroll
    for (int r = 0; r < 8; ++r) {
      float a = B2[wave * WB + lane + 32 * r];
      s += a * a;
    }
    na0 = sqrtf(wred(s));
  }

  // ---- P4: H1 = A0 @ W1 (K=256) ----
  gemm_tile(B2, WB, w1, HIDV, false, HIDV, B3, WB, wave * 16, lane);
  __syncthreads();

  // ---- P5: a1 = gelu(h1) -> B2, gp1 in place; ||a1|| ----
  for (int idx = tid; idx < 16 * HIDV; idx += 512) {
    int t = idx >> 8, j = idx & 255;
    float h = B3[t * WB + j];
    float a = gelu_f(h);
    B2[t * WB + j] = a;
    B3[t * WB + j] = gelu_g(h);
    A1f[(size_t)(T0 + t) * HIDV + j] = a;
  }
  __syncthreads();
  float na1;
  {
    float s = 0.f;
#pragma unroll
    for (int r = 0; r < 8; ++r) {
      float a = B2[wave * WB + lane + 32 * r];
      s += a * a;
    }
    na1 = sqrtf(wred(s));
  }

  // ---- P6: H2 = A1 @ W2 (16x64, K=256) ----
  if (wave < 4)
    gemm_tile(B2, WB, w2, DIMV, false, HIDV, sT, WS, wave * 16, lane);
  __syncthreads();

  // ---- P7: loss head: pred/dpred, g_nw, dh2 (+ scales for both) ----
  {
    const int t = wave;
    const size_t tok = (size_t)(T0 + t);
    float h0v = sT[t * WS + lane], h1v = sT[t * WS + lane + 32];
    float r = rsqrtf(wred(h0v * h0v + h1v * h1v) * (1.0f / 64.0f) + EPSV);
    float nw0 = norm_w[lane], nw1 = norm_w[lane + 32];
    float x0 = sX[t * WS + lane], x1 = sX[t * WS + lane + 32];
    float y0 = sY[t * WS + lane], y1 = sY[t * WS + lane + 32];
    float p0 = h0v * r * nw0 + x0, p1 = h1v * r * nw1 + x1;
    float dp0 = (p0 - y0) * (2.0f / 64.0f);
    float dp1 = (p1 - y1) * (2.0f / 64.0f);
    float S = wred(dp0 * h0v * nw0 + dp1 * h1v * nw1);
    // grad wrt norm weight
    float gn0 = dp0 * h0v * r, gn1 = dp1 * h1v * r;
    float snw = sc_scale(sqrtf(wred(gn0 * gn0 + gn1 * gn1)));
    GNf[tok * 64 + lane]      = -snw * gn0;
    GNf[tok * 64 + lane + 32] = -snw * gn1;
    // grad wrt h2 (rmsnorm backward)
    float k = r * r * r * (1.0f / 64.0f) * S;
    float d0 = dp0 * r * nw0 - k * h0v;
    float d1 = dp1 * r * nw1 - k * h1v;
    float nd2 = sqrtf(wred(d0 * d0 + d1 * d1));
    float s2 = sc_scale(na1 * nd2);
    D2f[tok * 64 + lane]      = -s2 * d0;
    D2f[tok * 64 + lane + 32] = -s2 * d1;
    sT[t * WS + lane] = d0;
    sT[t * WS + lane + 32] = d1;
  }
  __syncthreads();

  // ---- P8: dA1 = dh2 @ W2^T (16x256, K=64) ----
  gemm_tile(sT, WS, w2, DIMV, true, DIMV, B2, WB, wave * 16, lane);
  __syncthreads();

  // ---- P9: dH1 = dA1 * gp1; scale + write ----
  {
    const int t = wave;
    const size_t tok = (size_t)(T0 + t);
    float dj[8];
    float ss = 0.f;
#pragma unroll
    for (int r = 0; r < 8; ++r) {
      int j = lane + 32 * r;
      float d = B2[t * WB + j] * B3[t * WB + j];
      dj[r] = d;
      ss += d * d;
      B2[t * WB + j] = d;
    }
    float s1 = sc_scale(na0 * sqrtf(wred(ss)));
#pragma unroll
    for (int r = 0; r < 8; ++r)
      D1f[tok * HIDV + lane + 32 * r] = -s1 * dj[r];
  }
  __syncthreads();

  // ---- P10: dA0 = dH1 @ W1^T (K=256) ----
  gemm_tile(B2, WB, w1, HIDV, true, HIDV, B3, WB, wave * 16, lane);
  __syncthreads();

  // ---- P11: dH0 = dA0 * gp0; scale + write ----
  {
    const int t = wave;
    const size_t tok = (size_t)(T0 + t);
    float dj[8];
    float ss = 0.f;
#pragma unroll
    for (int r = 0; r < 8; ++r) {
      int j = lane + 32 * r;
      float d = B3[t * WB + j] * B1[t * WB + j];
      dj[r] = d;
      ss += d * d;
    }
    float s0 = sc_scale(nx * sqrtf(wred(ss)));
#pragma unroll
    for (int r = 0; r < 8; ++r)
      D0f[tok * HIDV + lane + 32 * r] = -s0 * dj[r];
  }
}

// ---------------------------------------------------------------------------
// Phase 2: gated scan over tokens, each thread owns 4 consecutive columns.
// Streams 404 MB with nontemporal global_store_b128; factor reads stay in L2.
// All segment boundaries (16384/81920/98304) are 1024-aligned, so no block
// and no thread straddles tensors; f2 vector loads are 16B-aligned.
// ---------------------------------------------------------------------------
__launch_bounds__(256)
__global__ void titans_phase2(const float* __restrict__ w0,
                              const float* __restrict__ w1,
                              const float* __restrict__ w2,
                              const float* __restrict__ norm_w,
                              const float* __restrict__ ws,
                              float* __restrict__ out) {
  const int col = (blockIdx.x * 256 + threadIdx.x) * 4;
  const int b = blockIdx.y;
  if (col >= DTOT) return;

  const float *f1, *f2;   // surprise(t) = f1[t*st1] * f2[t*st2 .. +3]
  int st1, st2;
  v4f acc;
  if (col < 16384) {                       // w0: x[i] * d0[j..j+3]
    int i = col >> 8, j = col & 255;
    f1 = ws + OFF_XF  + i; st1 = 64;
    f2 = ws + OFF_D0F + j; st2 = 256;
    acc = *(const v4f*)(w0 + col);
  } else if (col < 81920) {                // w1: a0[i] * d1[j..j+3]
    int e = col - 16384;
    int i = e >> 8, j = e & 255;
    f1 = ws + OFF_A0F + i; st1 = 256;
    f2 = ws + OFF_D1F + j; st2 = 256;
    acc = *(const v4f*)(w1 + e);
  } else if (col < 98304) {                // w2: a1[i] * d2[j..j+3]
    int e = col - 81920;
    int i = e >> 6, j = e & 63;
    f1 = ws + OFF_A1F + i; st1 = 256;
    f2 = ws + OFF_D2F + j; st2 = 64;
    acc = *(const v4f*)(w2 + e);
  } else {                                 // norm_w: 1.0 * gn[e..e+3]
    int e = col - 98304;
    f1 = ws + OFF_ONE;     st1 = 0;
    f2 = ws + OFF_GNF + e; st2 = 64;
    acc = *(const v4f*)(norm_w + e);
  }
  f1 += (size_t)b * NTOK * st1;
  f2 += (size_t)b * NTOK * st2;

  float* o = out + (size_t)b * NP1 * DTOT + col;
  __builtin_nontemporal_store(acc, (v4f*)o);   // scan step 0: prev state
  o += DTOT;
#pragma unroll 2
  for (int t = 0; t < NTOK; ++t) {
    float s1v = f1[(size_t)t * st1];
    v4f  s2v = *(const v4f*)(f2 + (size_t)t * st2);
#pragma unroll
    for (int q = 0; q < 4; ++q)                 // out_t = g*out_{t-1} + s_t
      acc[q] = fmaf(GATE, acc[q], s1v * s2v[q]);
    __builtin_nontemporal_store(acc, (v4f*)o);
    o += DTOT;
  }
}

extern "C" void kernel_launch(void* const* d_in, const int* in_sizes, int n_in,
                              void* d_out, int out_size, void* d_ws, size_t ws_size,
                              hipStream_t stream) {
  const float* seq      = (const float*)d_in[0];
  const float* w0       = (const float*)d_in[1];
  const float* w1       = (const float*)d_in[2];
  const float* w2       = (const float*)d_in[3];
  const float* norm_w   = (const float*)d_in[4];
  const float* store_nw = (const float*)d_in[5];
  float* ws  = (float*)d_ws;   // needs ~5 MB (OFF_ONE+1 floats)
  float* out = (float*)d_out;

  titans_phase1<<<dim3(64), dim3(512), 0, stream>>>(seq, w0, w1, w2, norm_w,
                                                    store_nw, ws);
  titans_phase2<<<dim3((DTOT + 1023) / 1024, BATCH), dim3(256), 0, stream>>>(
      w0, w1, w2, norm_w, ws, out);
}